// TfmrAttention_8392366096841
// MI455X (gfx1250) — compile-verified
//
#include <hip/hip_runtime.h>

typedef __attribute__((ext_vector_type(2))) float v2f;
typedef __attribute__((ext_vector_type(4))) float v4f;
typedef __attribute__((ext_vector_type(8))) float v8f;

#define B_   2
#define S_   2048
#define E_   1024
#define H_   16
#define D_   64
#define NQKV 3072
#define MASKED_BIAS (-10000.0f)

// D = A(16x4) * B(4x16) + C, fp32 tensor path. Per ISA layout:
//   lane = half*16 + lrow ; A lane holds A[lrow][k0], A[lrow][k0+1], k0 = kb + 2*half
//   B lane holds B[k0][lrow], B[k0+1][lrow]
//   C/D vgpr r, lane: C[r + 8*half][lrow]
__device__ __forceinline__ v8f wmma_f32(v2f a, v2f b, v8f c) {
  return __builtin_amdgcn_wmma_f32_16x16x4_f32(false, a, false, b, (short)0, c,
                                               false, false);
}

// ---------------------------------------------------------------------------
// Kernel 1: qkv = hs @ w_attn + b_attn, scattered to q/k/v in [B,H,S,D] layout
// grid: (3072/64, 4096/256), block 256 (8 waves; wave owns 32 rows x 64 cols =
// 2 M-tiles x 4 N-tiles; each B fragment feeds 2 WMMAs)
// ---------------------------------------------------------------------------
__global__ __launch_bounds__(256) void qkv_kernel(
    const float* __restrict__ hs, const float* __restrict__ w,
    const float* __restrict__ bias, float* __restrict__ qws,
    float* __restrict__ kws, float* __restrict__ vws) {
  const int wid = threadIdx.x >> 5;
  const int lane = threadIdx.x & 31;
  const int half = lane >> 4;
  const int lrow = lane & 15;
  const int n0 = blockIdx.x * 64;
  const int m0 = blockIdx.y * 256 + wid * 32;

  v8f acc[2][4] = {};
  const float* arow0 = hs + (size_t)(m0 + lrow) * E_;
  const float* arow1 = hs + (size_t)(m0 + 16 + lrow) * E_;
  for (int kb = 0; kb < E_; kb += 4) {
    const int k0 = kb + half * 2;
    v2f a0 = *(const v2f*)(arow0 + k0);
    v2f a1 = *(const v2f*)(arow1 + k0);
#pragma unroll
    for (int t = 0; t < 4; ++t) {
      const int col = n0 + t * 16 + lrow;
      v2f bf;
      bf.x = w[(size_t)k0 * NQKV + col];
      bf.y = w[(size_t)(k0 + 1) * NQKV + col];
      acc[0][t] = wmma_f32(a0, bf, acc[0][t]);
      acc[1][t] = wmma_f32(a1, bf, acc[1][t]);
    }
  }

  const int which = n0 >> 10;         // 0=q, 1=k, 2=v (n0 is 64-aligned)
  const int h = (n0 & 1023) >> 6;     // head (constant across the 64-col strip)
  float* dst = (which == 0) ? qws : ((which == 1) ? kws : vws);
#pragma unroll
  for (int mt = 0; mt < 2; ++mt) {
#pragma unroll
    for (int t = 0; t < 4; ++t) {
      const int d = t * 16 + lrow;
      const float bv = bias[n0 + t * 16 + lrow];
#pragma unroll
      for (int r = 0; r < 8; ++r) {
        const int m = m0 + mt * 16 + r + half * 8;
        const int bb = m >> 11;
        const int s = m & (S_ - 1);
        dst[(((size_t)bb * H_ + h) * S_ + s) * D_ + d] = acc[mt][t][r] + bv;
      }
    }
  }
}

// ---------------------------------------------------------------------------
// Kernel 2: scores = Q @ K^T / 8 with causal bias-mask, per (b,h)
// grid: (2048/64 k-tiles, 2048/128 q-tiles, 32 bh), block 256
// ---------------------------------------------------------------------------
__global__ __launch_bounds__(256) void scores_kernel(
    const float* __restrict__ qws, const float* __restrict__ kws,
    float* __restrict__ wts) {
  const int bh = blockIdx.z;
  const int kcol0 = blockIdx.x * 64;
  const int qblk = blockIdx.y * 128;
  float* Wout = wts + (size_t)bh * S_ * S_;

  if (kcol0 > qblk + 127) {  // tile fully above the diagonal: bias fill only
    const int r = threadIdx.x >> 1;
    const int c = (threadIdx.x & 1) * 32;
    float* p = Wout + (size_t)(qblk + r) * S_ + kcol0 + c;
    v4f mb = {MASKED_BIAS, MASKED_BIAS, MASKED_BIAS, MASKED_BIAS};
#pragma unroll
    for (int i = 0; i < 8; ++i) *(v4f*)(p + i * 4) = mb;
    return;
  }

  const int wid = threadIdx.x >> 5;
  const int lane = threadIdx.x & 31;
  const int half = lane >> 4;
  const int lrow = lane & 15;
  const int m0 = qblk + wid * 16;
  const float* Q = qws + (size_t)bh * S_ * D_;
  const float* Kt = kws + (size_t)bh * S_ * D_;

  v8f acc[4] = {};
  const float* qrow = Q + (size_t)(m0 + lrow) * D_;
#pragma unroll
  for (int kb = 0; kb < D_; kb += 4) {
    const int k0 = kb + half * 2;
    v2f a = *(const v2f*)(qrow + k0);
#pragma unroll
    for (int t = 0; t < 4; ++t) {
      // B[k][n] = K[n][k] -> contiguous float2 along D
      v2f bf = *(const v2f*)(Kt + (size_t)(kcol0 + t * 16 + lrow) * D_ + k0);
      acc[t] = wmma_f32(a, bf, acc[t]);
    }
  }
#pragma unroll
  for (int t = 0; t < 4; ++t) {
    const int kg = kcol0 + t * 16 + lrow;
#pragma unroll
    for (int r = 0; r < 8; ++r) {
      const int qg = m0 + r + half * 8;
      Wout[(size_t)qg * S_ + kg] =
          (kg <= qg) ? acc[t][r] * 0.125f : MASKED_BIAS;
    }
  }
}

// ---------------------------------------------------------------------------
// Kernel 3: row softmax in place over attn_weights (row length 2048)
// grid: 65536 blocks, block 256 (8 elems/thread held in registers)
// ---------------------------------------------------------------------------
__global__ __launch_bounds__(256) void softmax_kernel(float* __restrict__ wts) {
  __shared__ float red[256];
  float* Wrow = wts + (size_t)blockIdx.x * S_;
  const int tid = threadIdx.x;

  float vals[8];
  float lmax = -3.402823466e38f;
#pragma unroll
  for (int i = 0; i < 2; ++i) {
    v4f v = *(const v4f*)(Wrow + tid * 8 + i * 4);
#pragma unroll
    for (int j = 0; j < 4; ++j) {
      vals[i * 4 + j] = v[j];
      lmax = fmaxf(lmax, v[j]);
    }
  }
  red[tid] = lmax;
  __syncthreads();
  for (int s = 128; s > 0; s >>= 1) {
    if (tid < s) red[tid] = fmaxf(red[tid], red[tid + s]);
    __syncthreads();
  }
  const float rmax = red[0];
  __syncthreads();

  float lsum = 0.0f;
#pragma unroll
  for (int i = 0; i < 8; ++i) {
    vals[i] = __expf(vals[i] - rmax);
    lsum += vals[i];
  }
  red[tid] = lsum;
  __syncthreads();
  for (int s = 128; s > 0; s >>= 1) {
    if (tid < s) red[tid] += red[tid + s];
    __syncthreads();
  }
  const float inv = 1.0f / red[0];
#pragma unroll
  for (int i = 0; i < 2; ++i) {
    v4f v;
#pragma unroll
    for (int j = 0; j < 4; ++j) v[j] = vals[i * 4 + j] * inv;
    *(v4f*)(Wrow + tid * 8 + i * 4) = v;
  }
}

// ---------------------------------------------------------------------------
// Kernel 4: ctx = attn_weights @ V per (b,h); causal bound on k loop
// grid: (1, 2048/128, 32), block 256
// ---------------------------------------------------------------------------
__global__ __launch_bounds__(256) void av_kernel(
    const float* __restrict__ wts, const float* __restrict__ vws,
    float* __restrict__ ctx) {
  const int bh = blockIdx.z;
  const int wid = threadIdx.x >> 5;
  const int lane = threadIdx.x & 31;
  const int half = lane >> 4;
  const int lrow = lane & 15;
  const int qblk = blockIdx.y * 128;
  const int m0 = qblk + wid * 16;
  const float* Wt = wts + (size_t)bh * S_ * S_;
  const float* V = vws + (size_t)bh * S_ * D_;
  const int kmax = qblk + 128;  // weights are exactly 0 beyond each row's diag

  v8f acc[4] = {};
  const float* wrow = Wt + (size_t)(m0 + lrow) * S_;
  for (int kb = 0; kb < kmax; kb += 4) {
    const int k0 = kb + half * 2;
    v2f a = *(const v2f*)(wrow + k0);
#pragma unroll
    for (int t = 0; t < 4; ++t) {
      v2f bf;
      bf.x = V[(size_t)k0 * D_ + t * 16 + lrow];
      bf.y = V[(size_t)(k0 + 1) * D_ + t * 16 + lrow];
      acc[t] = wmma_f32(a, bf, acc[t]);
    }
  }
#pragma unroll
  for (int t = 0; t < 4; ++t) {
    const int d = t * 16 + lrow;
#pragma unroll
    for (int r = 0; r < 8; ++r) {
      const int s = m0 + r + half * 8;
      ctx[((size_t)bh * S_ + s) * D_ + d] = acc[t][r];
    }
  }
}

// ---------------------------------------------------------------------------
// Kernel 5: out = merge_heads(ctx) @ w_proj + b_proj
// grid: (1024/64, 4096/256), block 256 (wave = 32 rows x 64 cols)
// ---------------------------------------------------------------------------
__global__ __launch_bounds__(256) void proj_kernel(
    const float* __restrict__ ctx, const float* __restrict__ w,
    const float* __restrict__ bias, float* __restrict__ out) {
  const int wid = threadIdx.x >> 5;
  const int lane = threadIdx.x & 31;
  const int half = lane >> 4;
  const int lrow = lane & 15;
  const int n0 = blockIdx.x * 64;
  const int m0 = blockIdx.y * 256 + wid * 32;

  // A element (m, e): e -> (h = e/64, d = e%64), fetched from [B,H,S,D] ctx
  const int ma = m0 + lrow;
  const int mb2 = m0 + 16 + lrow;
  const float* abase0 =
      ctx + ((size_t)(ma >> 11) * H_) * S_ * D_ + (size_t)(ma & (S_ - 1)) * D_;
  const float* abase1 =
      ctx + ((size_t)(mb2 >> 11) * H_) * S_ * D_ + (size_t)(mb2 & (S_ - 1)) * D_;

  v8f acc[2][4] = {};
  for (int kb = 0; kb < E_; kb += 4) {
    const int k0 = kb + half * 2;              // even, so (k0, k0+1) same head
    const size_t aoff = (size_t)(k0 >> 6) * S_ * D_ + (k0 & 63);
    v2f a0 = *(const v2f*)(abase0 + aoff);
    v2f a1 = *(const v2f*)(abase1 + aoff);
#pragma unroll
    for (int t = 0; t < 4; ++t) {
      const int col = n0 + t * 16 + lrow;
      v2f bf;
      bf.x = w[(size_t)k0 * E_ + col];
      bf.y = w[(size_t)(k0 + 1) * E_ + col];
      acc[0][t] = wmma_f32(a0, bf, acc[0][t]);
      acc[1][t] = wmma_f32(a1, bf, acc[1][t]);
    }
  }
#pragma unroll
  for (int mt = 0; mt < 2; ++mt) {
#pragma unroll
    for (int t = 0; t < 4; ++t) {
      const int col = n0 + t * 16 + lrow;
      const float bv = bias[col];
#pragma unroll
      for (int r = 0; r < 8; ++r) {
        const int mg = m0 + mt * 16 + r + half * 8;
        out[(size_t)mg * E_ + col] = acc[mt][t][r] + bv;
      }
    }
  }
}

// ---------------------------------------------------------------------------
extern "C" void kernel_launch(void* const* d_in, const int* in_sizes, int n_in,
                              void* d_out, int out_size, void* d_ws,
                              size_t ws_size, hipStream_t stream) {
  (void)in_sizes; (void)n_in; (void)out_size; (void)ws_size;
  const float* hs     = (const float*)d_in[0];
  const float* w_attn = (const float*)d_in[1];
  const float* b_attn = (const float*)d_in[2];
  const float* w_proj = (const float*)d_in[3];
  const float* b_proj = (const float*)d_in[4];

  float* out = (float*)d_out;                       // [B,S,E] = 4Mi floats
  float* weights = out + (size_t)B_ * S_ * E_;      // [B,H,S,S] = 128Mi floats

  float* ws = (float*)d_ws;
  const size_t seg = (size_t)B_ * H_ * S_ * D_;     // 4Mi floats (16 MB)
  float* qws = ws;
  float* kws = ws + seg;
  float* vws = ws + 2 * seg;
  float* ctx = ws + 3 * seg;                        // total 64 MB workspace

  qkv_kernel<<<dim3(NQKV / 64, (B_ * S_) / 256), 256, 0, stream>>>(
      hs, w_attn, b_attn, qws, kws, vws);
  scores_kernel<<<dim3(S_ / 64, S_ / 128, B_ * H_), 256, 0, stream>>>(
      qws, kws, weights);
  softmax_kernel<<<dim3(B_ * H_ * S_), 256, 0, stream>>>(weights);
  av_kernel<<<dim3(1, S_ / 128, B_ * H_), 256, 0, stream>>>(weights, vws, ctx);
  proj_kernel<<<dim3(E_ / 64, (B_ * S_) / 256), 256, 0, stream>>>(
      ctx, w_proj, b_proj, out);
}